// ODE_51831665328625
// MI455X (gfx1250) — compile-verified
//
#include <hip/hip_runtime.h>
#include <math.h>

typedef float v2f __attribute__((ext_vector_type(2)));
typedef float v8f __attribute__((ext_vector_type(8)));

#define NB    8192
#define DIM   128
#define HID   64
#define RPB   32            // rows per block
#define TPB   256           // 8 waves (wave32)
#define NBLK  (NB / RPB)    // 256 blocks
#define NINTERVALS 11
#define MAXSTEPS   24

// LDS pitches (floats). Lane stride mod 64 banks == 4 -> conflict-free frag loads.
#define ZP   132   // 128-col row-major tiles
#define HP   68    // 64-col row-major tiles
#define W1P  68
#define W2P  68
#define W3P  132

// shared-memory layout (float offsets)
#define OFF_W1  0
#define OFF_W2  (OFF_W1 + 128*W1P)      // 8704
#define OFF_W3  (OFF_W2 + 64*W2P)       // 13056
#define OFF_B1  (OFF_W3 + 64*W3P)       // 21504
#define OFF_B2  (OFF_B1 + 64)
#define OFF_B3  (OFF_B2 + 64)
#define OFF_ZS  (OFF_B3 + 128)          // 21760 (also layer-3 output / k staging)
#define OFF_H1  (OFF_ZS + RPB*ZP)       // 25984
#define OFF_H2  (OFF_H1 + RPB*HP)       // 28160
#define OFF_RED (OFF_H2 + RPB*HP)       // 30336
#define SMEM_FLOATS (OFF_RED + 8)       // 30344
#define SMEM_BYTES  (SMEM_FLOATS * 4)   // 121376 B  (< 320KB WGP LDS)

// D = Z(RPBxK) x Wt(KxN) + bias, optional silu, written row-major to outs.
// One wave computes one 16x16 tile per 64 output columns (8 waves cover 32xN, N=64;
// for N=128 each wave does 2 tiles).
template<int K, int N, int ZPITCH, int WPITCH, int OPITCH, bool SILU>
__device__ __forceinline__ void mlp_gemm(const float* __restrict__ zs,
                                         const float* __restrict__ wt,
                                         const float* __restrict__ bias,
                                         float* __restrict__ outs,
                                         int lane, int wave)
{
  const int ln = lane & 15;
  const int hl = lane >> 4;            // 0: K+0/1, 1: K+2/3 (A/B frag halves)
  const int m0 = (wave >> 2) << 4;     // row-tile base (0 or 16)
#pragma unroll
  for (int nt = 0; nt < N / 64; ++nt) {
    const int n0 = (((wave & 3) + nt * 4) << 4);
    v8f acc = {};
#pragma unroll
    for (int kb = 0; kb < K; kb += 4) {
      const int ka = kb + 2 * hl;
      // A fragment: row m0+ln, K = ka, ka+1 (adjacent -> one b64 LDS load)
      const float2 av = *(const float2*)&zs[(m0 + ln) * ZPITCH + ka];
      v2f a; a.x = av.x; a.y = av.y;
      // B fragment: col n0+ln, K = ka, ka+1
      v2f b;
      b.x = wt[ka * WPITCH + n0 + ln];
      b.y = wt[(ka + 1) * WPITCH + n0 + ln];
      acc = __builtin_amdgcn_wmma_f32_16x16x4_f32(false, a, false, b,
                                                  (short)0, acc, false, false);
    }
    const float bv = bias[n0 + ln];
#pragma unroll
    for (int r = 0; r < 8; ++r) {
      float v = acc[r] + bv;
      if (SILU) v = v / (1.0f + expf(-v));
      outs[(m0 + r + 8 * hl) * OPITCH + n0 + ln] = v;
    }
  }
}

// One RK45 step attempt for a 32-row tile. scal: [0]=t_c [1]=h_c [2]=accept_pending [3]=err_sum
__global__ __launch_bounds__(TPB) void try_step_kernel(
    float* __restrict__ y_cur, float* __restrict__ y_cand,
    float* __restrict__ scal, const float* __restrict__ t_arr,
    const float* __restrict__ W1, const float* __restrict__ b1,
    const float* __restrict__ W2, const float* __restrict__ b2,
    const float* __restrict__ W3, const float* __restrict__ b3,
    int interval)
{
  extern __shared__ float sm[];
  float* W1t = sm + OFF_W1;
  float* W2t = sm + OFF_W2;
  float* W3t = sm + OFF_W3;
  float* b1s = sm + OFF_B1;
  float* b2s = sm + OFF_B2;
  float* b3s = sm + OFF_B3;
  float* zs  = sm + OFF_ZS;
  float* h1s = sm + OFF_H1;
  float* h2s = sm + OFF_H2;
  float* red = sm + OFF_RED;

  const int t    = threadIdx.x;
  const int lane = t & 31;
  const int wave = t >> 5;
  const int rw   = t >> 3;             // local row 0..31
  const int cb   = (t & 7) << 4;       // col base 0..112 (16 cols per thread)
  const size_t gbase = ((size_t)blockIdx.x * RPB + rw) * DIM + cb;

  const float tc    = scal[0];
  const float hc    = scal[1];
  const bool  acc_p = scal[2] != 0.0f;
  const float rem   = t_arr[interval + 1] - tc;

  // commit previously accepted candidate (idempotent), load y into regs
  float y[16];
  if (acc_p) {
#pragma unroll
    for (int i = 0; i < 16; i += 4) {
      float4 v = *(const float4*)&y_cand[gbase + i];
      *(float4*)&y[i] = v;
      *(float4*)&y_cur[gbase + i] = v;
    }
  } else {
#pragma unroll
    for (int i = 0; i < 16; i += 4)
      *(float4*)&y[i] = *(const float4*)&y_cur[gbase + i];
  }

  if (rem <= 1e-9f) return;            // uniform exit: interval already done

  const float h = fmaxf(fminf(hc, rem), 1e-8f);

  // weights -> LDS, transposed + pitch-padded
  for (int e = t; e < HID * DIM; e += TPB) {     // W1 (64x128) -> W1t[k][n]
    int n = e / DIM, k = e % DIM;
    W1t[k * W1P + n] = W1[e];
  }
  for (int e = t; e < HID * HID; e += TPB) {     // W2 (64x64) -> W2t[k][n]
    int n = e / HID, k = e % HID;
    W2t[k * W2P + n] = W2[e];
  }
  for (int e = t; e < DIM * HID; e += TPB) {     // W3 (128x64) -> W3t[k][n]
    int n = e / HID, k = e % HID;
    W3t[k * W3P + n] = W3[e];
  }
  if (t < 64)  { b1s[t] = b1[t]; b2s[t] = b2[t]; }
  if (t < 128) { b3s[t] = b3[t]; }

  float k1[16], k2[16], k3[16], k4[16], k5[16], z[16], yn[16], ea[16];

  auto feval = [&](const float* zin, float* kout) {
#pragma unroll
    for (int i = 0; i < 16; i += 4)
      *(float4*)&zs[rw * ZP + cb + i] = *(const float4*)&zin[i];
    __syncthreads();
    mlp_gemm<DIM, HID, ZP, W1P, HP, true >(zs,  W1t, b1s, h1s, lane, wave);
    __syncthreads();
    mlp_gemm<HID, HID, HP, W2P, HP, true >(h1s, W2t, b2s, h2s, lane, wave);
    __syncthreads();
    mlp_gemm<HID, DIM, HP, W3P, ZP, false>(h2s, W3t, b3s, zs,  lane, wave);
    __syncthreads();
#pragma unroll
    for (int i = 0; i < 16; i += 4)
      *(float4*)&kout[i] = *(const float4*)&zs[rw * ZP + cb + i];
  };

  feval(y, k1);
#pragma unroll
  for (int i = 0; i < 16; ++i) z[i] = y[i] + h * (0.2f * k1[i]);
  feval(z, k2);
#pragma unroll
  for (int i = 0; i < 16; ++i)
    z[i] = y[i] + h * ((3.0f/40.0f) * k1[i] + (9.0f/40.0f) * k2[i]);
  feval(z, k3);
#pragma unroll
  for (int i = 0; i < 16; ++i)
    z[i] = y[i] + h * ((44.0f/45.0f) * k1[i] + (-56.0f/15.0f) * k2[i] + (32.0f/9.0f) * k3[i]);
  feval(z, k4);
#pragma unroll
  for (int i = 0; i < 16; ++i)
    z[i] = y[i] + h * ((19372.0f/6561.0f) * k1[i] + (-25360.0f/2187.0f) * k2[i]
                     + (64448.0f/6561.0f) * k3[i] + (-212.0f/729.0f) * k4[i]);
  feval(z, k5);
#pragma unroll
  for (int i = 0; i < 16; ++i)
    z[i] = y[i] + h * ((9017.0f/3168.0f) * k1[i] + (-355.0f/33.0f) * k2[i]
                     + (46732.0f/5247.0f) * k3[i] + (49.0f/176.0f) * k4[i]
                     + (-5103.0f/18656.0f) * k5[i]);
  feval(z, k2);                         // k2 := k6 (k2's b/e coefficients are 0)
#pragma unroll
  for (int i = 0; i < 16; ++i)
    yn[i] = y[i] + h * ((35.0f/384.0f) * k1[i] + (500.0f/1113.0f) * k3[i]
                      + (125.0f/192.0f) * k4[i] + (-2187.0f/6784.0f) * k5[i]
                      + (11.0f/84.0f) * k2[i]);
  const float e1 = 35.0f/384.0f   - 5179.0f/57600.0f;
  const float e3 = 500.0f/1113.0f - 7571.0f/16695.0f;
  const float e4 = 125.0f/192.0f  - 393.0f/640.0f;
  const float e5 = -2187.0f/6784.0f + 92097.0f/339200.0f;
  const float e6 = 11.0f/84.0f    - 187.0f/2100.0f;
#pragma unroll
  for (int i = 0; i < 16; ++i)
    ea[i] = e1 * k1[i] + e3 * k3[i] + e4 * k4[i] + e5 * k5[i] + e6 * k2[i];
  feval(yn, k1);                        // k1 := k7

  float loc = 0.0f;
#pragma unroll
  for (int i = 0; i < 16; ++i) {
    const float err = h * (ea[i] + (-1.0f/40.0f) * k1[i]);
    const float sc  = 1e-6f + 1e-5f * fmaxf(fabsf(y[i]), fabsf(yn[i]));
    const float r   = err / sc;
    loc += r * r;
  }
#pragma unroll
  for (int i = 0; i < 16; i += 4)
    *(float4*)&y_cand[gbase + i] = *(const float4*)&yn[i];

  // block reduction of squared error contributions
  for (int off = 16; off > 0; off >>= 1) loc += __shfl_down(loc, off, 32);
  if (lane == 0) red[wave] = loc;
  __syncthreads();
  if (t == 0) {
    float s = 0.0f;
#pragma unroll
    for (int i = 0; i < 8; ++i) s += red[i];
    atomicAdd(&scal[3], s);
  }
}

__global__ void decide_kernel(float* __restrict__ scal,
                              const float* __restrict__ t_arr, int interval)
{
  const float tc  = scal[0];
  const float hc  = scal[1];
  const float rem = t_arr[interval + 1] - tc;
  if (rem <= 1e-9f) {
    scal[2] = 0.0f;                     // done: freeze t, h, y
  } else {
    const float h  = fmaxf(fminf(hc, rem), 1e-8f);
    const float en = sqrtf(scal[3] / ((float)NB * (float)DIM));
    const bool accept = en <= 1.0f;
    float fac = 0.9f * powf(en + 1e-16f, -0.2f);
    fac = fminf(fmaxf(fac, 0.2f), 10.0f);
    const float hn = fminf(fmaxf(h * fac, 1e-8f), 1.0f);
    if (accept) scal[0] = tc + h;
    scal[1] = hn;
    scal[2] = accept ? 1.0f : 0.0f;
  }
  scal[3] = 0.0f;
}

// Commit pending candidate + scatter ys[j] to its slot in d_out (xt | xt_rand).
__global__ void emit_kernel(const float* __restrict__ x0,  // non-null only for j==0
                            float* __restrict__ y_cur, const float* __restrict__ y_cand,
                            const float* __restrict__ scal, float* __restrict__ out,
                            const int* __restrict__ idx_actual,
                            const int* __restrict__ idx_rand, int j)
{
  int slot = -1;
  for (int k = 0; k < 10; ++k) if (idx_actual[k] == j) slot = k;
  for (int k = 0; k < 2;  ++k) if (idx_rand[k]   == j) slot = 10 + k;
  if (slot < 0) return;
  const size_t base = (size_t)slot * NB * DIM;
  const bool acc = scal[2] != 0.0f;
  const size_t idx = (size_t)blockIdx.x * blockDim.x + threadIdx.x;
  if (idx >= (size_t)NB * DIM) return;
  float v;
  if (x0) v = x0[idx];
  else    v = acc ? y_cand[idx] : y_cur[idx];
  y_cur[idx]     = v;
  out[base + idx] = v;
}

__global__ void init_kernel(float* __restrict__ scal, const float* __restrict__ t_arr)
{
  scal[0] = t_arr[0];
  scal[1] = 0.05f;      // H0
  scal[2] = 0.0f;
  scal[3] = 0.0f;
}

__global__ void advance_kernel(float* __restrict__ scal,
                               const float* __restrict__ t_arr, int next_start)
{
  scal[0] = t_arr[next_start];  // reset t to next interval start; h carries over
  scal[2] = 0.0f;               // candidate already committed by emit
  scal[3] = 0.0f;
}

extern "C" void kernel_launch(void* const* d_in, const int* in_sizes, int n_in,
                              void* d_out, int out_size, void* d_ws, size_t ws_size,
                              hipStream_t stream) {
  const float* x     = (const float*)d_in[0];
  const float* W1    = (const float*)d_in[1];
  const float* b1    = (const float*)d_in[2];
  const float* W2    = (const float*)d_in[3];
  const float* b2    = (const float*)d_in[4];
  const float* W3    = (const float*)d_in[5];
  const float* b3    = (const float*)d_in[6];
  const float* t_arr = (const float*)d_in[7];
  const int*   idxa  = (const int*)d_in[8];
  const int*   idxr  = (const int*)d_in[9];
  float* out = (float*)d_out;

  float* ws     = (float*)d_ws;
  float* scal   = ws;                       // 64 floats of scalar state
  float* y_cur  = ws + 64;                  // NB*DIM
  float* y_cand = y_cur + (size_t)NB * DIM; // NB*DIM

  (void)hipFuncSetAttribute((const void*)try_step_kernel,
                            hipFuncAttributeMaxDynamicSharedMemorySize, SMEM_BYTES);

  const int emit_blocks = (NB * DIM) / TPB; // 4096

  init_kernel<<<1, 1, 0, stream>>>(scal, t_arr);
  emit_kernel<<<emit_blocks, TPB, 0, stream>>>(x, y_cur, y_cand, scal, out, idxa, idxr, 0);

  for (int i = 0; i < NINTERVALS; ++i) {
    for (int s = 0; s < MAXSTEPS; ++s) {
      try_step_kernel<<<NBLK, TPB, SMEM_BYTES, stream>>>(
          y_cur, y_cand, scal, t_arr, W1, b1, W2, b2, W3, b3, i);
      decide_kernel<<<1, 1, 0, stream>>>(scal, t_arr, i);
    }
    emit_kernel<<<emit_blocks, TPB, 0, stream>>>(nullptr, y_cur, y_cand, scal, out,
                                                 idxa, idxr, i + 1);
    advance_kernel<<<1, 1, 0, stream>>>(scal, t_arr, i + 1);
  }
}